// Loss_39170101740021
// MI455X (gfx1250) — compile-verified
//
#include <hip/hip_runtime.h>
#include <hip/hip_bf16.h>

typedef __attribute__((ext_vector_type(2))) float v2f;
typedef __attribute__((ext_vector_type(8))) float v8f;

#define FLT_BIG 3.402823466e+38f
#define ROWT 4   // 16-row tiles per wave -> 64 rows

__device__ __forceinline__ unsigned int umin32(unsigned int a, unsigned int b) {
    return a < b ? a : b;
}

// Set first n uints of buffer to FLT_MAX bit pattern.
__global__ void chamfer_init_kernel(unsigned int* __restrict__ p, int n) {
    int i = blockIdx.x * blockDim.x + threadIdx.x;
    if (i < n) p[i] = 0x7F7FFFFFu;
}

// One wave owns a 64-row tile of x (four 16-row A/C fragment pairs) and a
// 1/csplit slice of gt's columns. K=4 fusion puts the FULL squared distance
// in the matrix pipe:
//   A = (x0, x1 | x2, 1),  B = (-2y0, -2y1 | -2y2, |y|^2),  C = |x_row|^2
// A and C are loop-invariant; each iteration: 3 coalesced loads + bfrag build
// feed FOUR back-to-back WMMAs. Outputs are clamped to [0, FLT_MAX] with one
// v_med3 each (canonical result, no sNaN-quieting max), then ALL min tracking
// runs in the unsigned-int domain (float order == uint order for values >= 0):
// v_min_u32 / v_min3_u32 carry no float canonicalization semantics. One
// full-wave atomicMin covers all 64 rows per column tile. Deterministic:
// min is order-independent.
__global__ void __launch_bounds__(256)
chamfer_pass_kernel(const float* __restrict__ x,      // [B, Nx, 3]
                    const float* __restrict__ gt,     // [B, 3, Ng]
                    unsigned int* __restrict__ minx,  // [B*Nx]  pre-init FLT_MAX
                    unsigned int* __restrict__ miny,  // [B*Ng]  pre-init FLT_MAX
                    int Nx, int Ng, int csplit)
{
    const int lane = threadIdx.x & 31;
    const int wid  = blockIdx.x * (blockDim.x >> 5) + (threadIdx.x >> 5);
    const int rowTiles = Nx >> 6;            // 64 rows per wave
    const int slice = wid % csplit;
    const int rb    = wid / csplit;
    const int b     = rb / rowTiles;
    const int rt    = rb % rowTiles;
    const int col   = lane & 15;
    const int hw    = lane >> 4;             // half-wave selects K pair {0,1}/{2,3}

    // ---- loop-invariant A and C fragments (four 16-row tiles) ----
    const int row0 = (rt << 6) + col;
    const float* xp = x + (size_t)(b * Nx + row0) * 3;

    v2f afrag[ROWT];
    v8f cfrag[ROWT];
    unsigned int rowminU[ROWT][8];
#pragma unroll
    for (int j = 0; j < ROWT; ++j) {
        const float x0 = xp[j * 48 + 0];     // +16 rows = +48 floats
        const float x1 = xp[j * 48 + 1];
        const float x2 = xp[j * 48 + 2];
        const float xx = x0 * x0 + x1 * x1 + x2 * x2;
        afrag[j].x = hw ? x2   : x0;
        afrag[j].y = hw ? 1.0f : x1;
#pragma unroll
        for (int r = 0; r < 8; ++r) {
            cfrag[j][r]    = __shfl(xx, r + 8 * hw, 32); // |x|^2 of row base+r+8*hw
            rowminU[j][r]  = 0x7F7FFFFFu;
        }
    }

    const int Nloc  = Ng / csplit;
    const int cbase = slice * Nloc;
    const float* g0 = gt + (size_t)(b * 3 + 0) * Ng + cbase;
    const float* g1 = g0 + Ng;
    const float* g2 = g1 + Ng;
    unsigned int* minyb = miny + (size_t)b * Ng + cbase;

    for (int t = 0; t < Nloc; t += 16) {
        const int n = t + col;
        const float y0 = g0[n], y1 = g1[n], y2 = g2[n];
        const float yy = y0 * y0 + y1 * y1 + y2 * y2;

        v2f bfrag;
        bfrag.x = hw ? (-2.0f * y2) : (-2.0f * y0);
        bfrag.y = hw ? yy           : (-2.0f * y1);

        // Four 16x16 distance tiles sharing one B fragment.
        v8f d[ROWT];
#pragma unroll
        for (int j = 0; j < ROWT; ++j)
            d[j] = __builtin_amdgcn_wmma_f32_16x16x4_f32(
                false, afrag[j], false, bfrag, (short)0, cfrag[j], false, false);

        unsigned int cminU = 0x7F7FFFFFu;
#pragma unroll
        for (int j = 0; j < ROWT; ++j) {
#pragma unroll
            for (int r = 0; r < 8; ++r) {
                const unsigned int dvU = __float_as_uint(
                    __builtin_amdgcn_fmed3f(d[j][r], 0.0f, FLT_BIG));
                rowminU[j][r] = umin32(rowminU[j][r], dvU);
                cminU         = umin32(cminU, dvU);
            }
        }
        // one atomic covers all 64 rows; lanes l and l^16 share an address.
        atomicMin(&minyb[n], cminU);
    }

    // ---- row mins: full-wave atomic merge (16 lanes per address) ----
    unsigned int* minxb = minx + (size_t)b * Nx + (rt << 6);
#pragma unroll
    for (int j = 0; j < ROWT; ++j)
#pragma unroll
        for (int r = 0; r < 8; ++r)
            atomicMin(&minxb[j * 16 + r + 8 * hw], rowminU[j][r]);
}

__global__ void chamfer_finalize_kernel(const float* __restrict__ minx_f,
                                        const float* __restrict__ minx_c,
                                        const float* __restrict__ miny_f,
                                        const float* __restrict__ miny_c,
                                        const float* __restrict__ alpha,
                                        float* __restrict__ out,
                                        int B, int Nf, int Nc, int Ng)
{
    __shared__ float s0[256], s1[256], s2[256], s3[256];
    const int tid = threadIdx.x;
    float a0 = 0.f, a1 = 0.f, a2 = 0.f, a3 = 0.f;
    for (int i = tid; i < B * Nf; i += 256) a0 += minx_f[i];
    for (int i = tid; i < B * Nc; i += 256) a1 += minx_c[i];
    for (int i = tid; i < B * Ng; i += 256) { a2 += miny_f[i]; a3 += miny_c[i]; }
    s0[tid] = a0; s1[tid] = a1; s2[tid] = a2; s3[tid] = a3;
    __syncthreads();
    for (int off = 128; off > 0; off >>= 1) {
        if (tid < off) {
            s0[tid] += s0[tid + off];
            s1[tid] += s1[tid + off];
            s2[tid] += s2[tid + off];
            s3[tid] += s3[tid + off];
        }
        __syncthreads();
    }
    if (tid == 0) {
        // equal-sized batches: mean_b(mean_n(.)) == flat mean
        float loss_fine   = s0[0] / (float)(B * Nf) + s2[0] / (float)(B * Ng);
        float loss_coarse = s1[0] / (float)(B * Nc) + s3[0] / (float)(B * Ng);
        float loss = loss_coarse + alpha[0] * loss_fine;
        out[0] = loss;
        out[1] = loss_coarse;
        out[2] = loss_fine;
    }
}

extern "C" void kernel_launch(void* const* d_in, const int* in_sizes, int n_in,
                              void* d_out, int out_size, void* d_ws, size_t ws_size,
                              hipStream_t stream) {
    const float* coarse = (const float*)d_in[0];
    const float* fine   = (const float*)d_in[1];
    const float* gt     = (const float*)d_in[2];
    const float* alpha  = (const float*)d_in[3];

    const int B  = 8;
    const int Nc = in_sizes[0] / (B * 3);
    const int Nf = in_sizes[1] / (B * 3);
    const int Ng = in_sizes[2] / (B * 3);

    // Workspace layout (contiguous, all init'd to FLT_MAX bits)
    unsigned int* minx_f = (unsigned int*)d_ws;              // B*Nf
    unsigned int* minx_c = minx_f + (size_t)B * Nf;          // B*Nc
    unsigned int* miny_f = minx_c + (size_t)B * Nc;          // B*Ng
    unsigned int* miny_c = miny_f + (size_t)B * Ng;          // B*Ng

    const int initN = B * (Nf + Nc + 2 * Ng);
    chamfer_init_kernel<<<(initN + 255) / 256, 256, 0, stream>>>(minx_f, initN);

    // 64 rows/wave; column-split for occupancy:
    //   fine:   8*(8192/64)*8  = 8192 waves -> 1024 blocks, 1024-col slices
    //   coarse: 8*(1024/64)*32 = 4096 waves ->  512 blocks,  256-col slices
    const int csF = 8, csC = 32;
    const int wavesF = B * (Nf / 64) * csF;
    const int wavesC = B * (Nc / 64) * csC;
    chamfer_pass_kernel<<<wavesF / 8, 256, 0, stream>>>(fine,   gt, minx_f, miny_f, Nf, Ng, csF);
    chamfer_pass_kernel<<<wavesC / 8, 256, 0, stream>>>(coarse, gt, minx_c, miny_c, Nc, Ng, csC);

    chamfer_finalize_kernel<<<1, 256, 0, stream>>>(
        (const float*)minx_f, (const float*)minx_c,
        (const float*)miny_f, (const float*)miny_c,
        alpha, (float*)d_out, B, Nf, Nc, Ng);
}